// MultiHeadAttention_41618233098569
// MI455X (gfx1250) — compile-verified
//
#include <hip/hip_runtime.h>
#include <math.h>

#define NB   16
#define NH_  16
#define T0_  4096
#define DQK_ 128
#define DV_  128
#define DIN  2048
#define NHD  (NH_ * DQK_)      // 2048 features per projection
#define SPLIT 8
#define NWAVE 8                // waves per block (256 threads)

typedef float v2f __attribute__((ext_vector_type(2)));
typedef float v8f __attribute__((ext_vector_type(8)));

// ---------------------------------------------------------------------------
// Kernel A: QKV projection.  out[m][j] = sum_k x[m][k] * W[j][k]
// One wave -> one 16(batch) x 16(feature) tile, K-loop via V_WMMA_F32_16X16X4_F32.
// A (16x4 f32): lanes 0-15 hold K=0,1 ; lanes 16-31 hold K=2,3  (per lane v2f)
// B (4x16 f32): lane n = column; lanes 0-15 rows K=0,1 ; lanes 16-31 rows K=2,3
// ---------------------------------------------------------------------------
__global__ __launch_bounds__(256) void qkv_proj_kernel(
    const float* __restrict__ x,
    const float* __restrict__ wq, const float* __restrict__ wk,
    const float* __restrict__ wv,
    float* __restrict__ qbuf, float* __restrict__ kbuf, float* __restrict__ vbuf) {
  const int wave = threadIdx.x >> 5;
  const int lane = threadIdx.x & 31;
  const int tile = blockIdx.x * NWAVE + wave;   // 0..383 (3 weights * 128 tiles)
  const int wid  = tile >> 7;
  const int j0   = (tile & 127) * 16;

  const float* __restrict__ W   = (wid == 0) ? wq : (wid == 1) ? wk : wv;
  float* __restrict__       out = (wid == 0) ? qbuf : (wid == 1) ? kbuf : vbuf;

  const int col  = lane & 15;            // matrix row (A) / column (B)
  const int koff = (lane >> 4) << 1;     // 0 for lanes 0-15, 2 for lanes 16-31

  const float* ap = x + (size_t)col * DIN + koff;             // x[m=col][k]
  const float* bp = W + (size_t)(j0 + col) * DIN + koff;      // W[j][k]

  v8f c = {};
#pragma unroll 4
  for (int k0 = 0; k0 < DIN; k0 += 4) {
    v2f a = *(const v2f*)(ap + k0);
    v2f b = *(const v2f*)(bp + k0);
    c = __builtin_amdgcn_wmma_f32_16x16x4_f32(false, a, false, b, (short)0, c,
                                              false, false);
  }

  const int mbase = (lane >> 4) << 3;    // C rows: r + 8*(lane/16)
#pragma unroll
  for (int r = 0; r < 8; ++r) {
    out[(size_t)(mbase + r) * NHD + j0 + col] = c[r];
  }
}

// ---------------------------------------------------------------------------
// Kernel B: flash-decode partial attention over one chunk of keys.
// grid = (N*H, SPLIT), block = 256 (8 waves).  Each wave: 1 key / iteration,
// lane holds dims 4*lane..4*lane+3 of k/v rows. Online softmax per wave,
// wave partials merged in LDS, chunk partial written to workspace.
// ---------------------------------------------------------------------------
__global__ __launch_bounds__(256) void attn_partial_kernel(
    const float* __restrict__ kc, const float* __restrict__ vc,
    const float* __restrict__ qbuf, const float* __restrict__ knew,
    const float* __restrict__ vnew, float* __restrict__ part) {
  const int nh    = blockIdx.x;        // 0..255
  const int chunk = blockIdx.y;        // 0..SPLIT-1
  const int n     = nh >> 4;
  const int h     = nh & 15;
  const int wave  = threadIdx.x >> 5;
  const int lane  = threadIdx.x & 31;

  const float scale = 1.0f / sqrtf((float)DQK_);
  const float* qp = qbuf + (size_t)n * NHD + h * DQK_ + lane * 4;
  float q0 = qp[0] * scale, q1 = qp[1] * scale, q2 = qp[2] * scale,
        q3 = qp[3] * scale;

  const int TTOT  = T0_ + 1;
  const int CHUNK = (TTOT + SPLIT - 1) / SPLIT;   // 513
  const int s0 = chunk * CHUNK;
  const int s1 = (s0 + CHUNK < TTOT) ? (s0 + CHUNK) : TTOT;

  const float* kbase = kc + (size_t)nh * T0_ * DQK_;
  const float* vbase = vc + (size_t)nh * T0_ * DV_;
  const float* knp   = knew + (size_t)n * NHD + h * DQK_;
  const float* vnp   = vnew + (size_t)n * NHD + h * DV_;

  float m = -__builtin_inff();
  float l = 0.f;
  float a0 = 0.f, a1 = 0.f, a2 = 0.f, a3 = 0.f;

  for (int s = s0 + wave; s < s1; s += NWAVE) {
    // prefetch ahead in the stream (speculative, lowers to global_prefetch_b8)
    if (s + 2 * NWAVE < T0_) {
      __builtin_prefetch(kbase + (size_t)(s + 2 * NWAVE) * DQK_ + lane * 4, 0, 1);
      __builtin_prefetch(vbase + (size_t)(s + 2 * NWAVE) * DV_ + lane * 4, 0, 1);
    }
    const float* kp = (s < T0_) ? (kbase + (size_t)s * DQK_ + lane * 4)
                                : (knp + lane * 4);
    float4 kv = *(const float4*)kp;
    float dot = q0 * kv.x + q1 * kv.y + q2 * kv.z + q3 * kv.w;
#pragma unroll
    for (int off = 16; off >= 1; off >>= 1) dot += __shfl_xor(dot, off, 32);

    const float* vp = (s < T0_) ? (vbase + (size_t)s * DV_ + lane * 4)
                                : (vnp + lane * 4);
    float4 vv = *(const float4*)vp;

    float mnew = fmaxf(m, dot);
    float f = __expf(m - mnew);
    float p = __expf(dot - mnew);
    a0 = a0 * f + p * vv.x;
    a1 = a1 * f + p * vv.y;
    a2 = a2 * f + p * vv.z;
    a3 = a3 * f + p * vv.w;
    l = l * f + p;
    m = mnew;
  }

  __shared__ float sm[NWAVE];
  __shared__ float sl[NWAVE];
  __shared__ float sacc[NWAVE][DV_];
  if (lane == 0) { sm[wave] = m; sl[wave] = l; }
  sacc[wave][lane * 4 + 0] = a0;
  sacc[wave][lane * 4 + 1] = a1;
  sacc[wave][lane * 4 + 2] = a2;
  sacc[wave][lane * 4 + 3] = a3;
  __syncthreads();

  if (wave == 0) {
    float mt = sm[0];
#pragma unroll
    for (int w = 1; w < NWAVE; ++w) mt = fmaxf(mt, sm[w]);
    float lt = 0.f, r0 = 0.f, r1 = 0.f, r2 = 0.f, r3 = 0.f;
#pragma unroll
    for (int w = 0; w < NWAVE; ++w) {
      float f = __expf(sm[w] - mt);
      lt += f * sl[w];
      r0 += f * sacc[w][lane * 4 + 0];
      r1 += f * sacc[w][lane * 4 + 1];
      r2 += f * sacc[w][lane * 4 + 2];
      r3 += f * sacc[w][lane * 4 + 3];
    }
    float* p = part + ((size_t)nh * SPLIT + chunk) * (DV_ + 2);
    if (lane == 0) { p[0] = mt; p[1] = lt; }
    p[2 + lane * 4 + 0] = r0;
    p[2 + lane * 4 + 1] = r1;
    p[2 + lane * 4 + 2] = r2;
    p[2 + lane * 4 + 3] = r3;
  }
}

// ---------------------------------------------------------------------------
// Kernel C: combine SPLIT chunk partials -> y[n][h*128+d]
// ---------------------------------------------------------------------------
__global__ __launch_bounds__(128) void attn_reduce_kernel(
    const float* __restrict__ part, float* __restrict__ y) {
  const int nh = blockIdx.x;
  const int n = nh >> 4, h = nh & 15;
  const int d = threadIdx.x;

  const float* base = part + (size_t)nh * SPLIT * (DV_ + 2);
  float mt = -__builtin_inff();
#pragma unroll
  for (int c = 0; c < SPLIT; ++c) mt = fmaxf(mt, base[(size_t)c * (DV_ + 2)]);
  float den = 0.f, num = 0.f;
#pragma unroll
  for (int c = 0; c < SPLIT; ++c) {
    const float* p = base + (size_t)c * (DV_ + 2);
    float f = __expf(p[0] - mt);
    den += f * p[1];
    num += f * p[2 + d];
  }
  y[(size_t)n * NHD + h * DV_ + d] = num / den;
}

// ---------------------------------------------------------------------------
// Kernel D: output projection.  out[m][j] = sum_k y[m][k] * w_o[k][j]
// Same WMMA tiling; B is row-major in j so per-lane K-strided loads.
// ---------------------------------------------------------------------------
__global__ __launch_bounds__(256) void out_proj_kernel(
    const float* __restrict__ y, const float* __restrict__ wo,
    float* __restrict__ out) {
  const int wave = threadIdx.x >> 5;
  const int lane = threadIdx.x & 31;
  const int tile = blockIdx.x * NWAVE + wave;   // 0..127
  const int j0   = tile * 16;

  const int col  = lane & 15;
  const int koff = (lane >> 4) << 1;

  const float* ap = y + (size_t)col * NHD + koff;

  v8f c = {};
#pragma unroll 4
  for (int k0 = 0; k0 < NHD; k0 += 4) {
    v2f a = *(const v2f*)(ap + k0);
    v2f b;
    b.x = wo[(size_t)(k0 + koff + 0) * DIN + j0 + col];
    b.y = wo[(size_t)(k0 + koff + 1) * DIN + j0 + col];
    c = __builtin_amdgcn_wmma_f32_16x16x4_f32(false, a, false, b, (short)0, c,
                                              false, false);
  }

  const int mbase = (lane >> 4) << 3;
#pragma unroll
  for (int r = 0; r < 8; ++r) {
    out[(size_t)(mbase + r) * DIN + j0 + col] = c[r];
  }
}

// ---------------------------------------------------------------------------
extern "C" void kernel_launch(void* const* d_in, const int* in_sizes, int n_in,
                              void* d_out, int out_size, void* d_ws,
                              size_t ws_size, hipStream_t stream) {
  const float* x  = (const float*)d_in[0];   // [16, 1, 2048]
  const float* kc = (const float*)d_in[1];   // [16, 16, 4096, 128]
  const float* vc = (const float*)d_in[2];   // [16, 16, 4096, 128]
  const float* wq = (const float*)d_in[3];   // [16, 128, 2048]
  const float* wk = (const float*)d_in[4];
  const float* wv = (const float*)d_in[5];
  const float* wo = (const float*)d_in[6];   // [2048, 2048]
  float* out = (float*)d_out;                // [16, 1, 2048]

  float* ws   = (float*)d_ws;
  float* qbuf = ws;                          // 16*2048
  float* kbuf = ws + 32768;                  // 16*2048
  float* vbuf = ws + 65536;                  // 16*2048
  float* ybuf = ws + 98304;                  // 16*2048
  float* part = ws + 131072;                 // 256*8*130

  // A: 384 tiles (3 weights x 128 feature tiles), 8 waves/block -> 48 blocks
  qkv_proj_kernel<<<48, 256, 0, stream>>>(x, wq, wk, wv, qbuf, kbuf, vbuf);
  // B: 256 (n,h) x 8 chunks = 2048 CTAs streaming 1 GB of KV
  attn_partial_kernel<<<dim3(256, SPLIT), 256, 0, stream>>>(kc, vc, qbuf, kbuf,
                                                            vbuf, part);
  // C: combine partials
  attn_reduce_kernel<<<256, 128, 0, stream>>>(part, ybuf);
  // D: 128 tiles, 8 waves/block -> 16 blocks
  out_proj_kernel<<<16, 256, 0, stream>>>(ybuf, wo, out);
}